// ATTLayer_66632122630368
// MI455X (gfx1250) — compile-verified
//
#include <hip/hip_runtime.h>

#define D 128
#define E 128
#define BATCH 4096

typedef __attribute__((ext_vector_type(2))) float v2f;
typedef __attribute__((ext_vector_type(8))) float v8f;

// workspace layout in floats
#define WS_G  0         // [128,128]  G = W^T W
#define WS_U  16384     // [128]      u = W^T b
#define WS_BB 16512     // [1]        b.b
#define WS_C  16640     // [4096]     c[b]
#define WS_Q  32768     // [4096,128] q[b] = G x[b] + u

// ---------------------------------------------------------------------------
// G = W^T W  (128x128x128 GEMM), one wave per 16x16 tile, WMMA f32 16x16x4.
// A[m][kk] = W[f0+kk][d0+m], B[kk][n] = W[f0+kk][n0+n].
// f32 WMMA lane layout: m/n = lane&15, kh = lane>>4; a[v] = A[m][2*kh+v].
// ---------------------------------------------------------------------------
__global__ void k_gram(const float* __restrict__ W, float* __restrict__ G) {
    const int lane = threadIdx.x;
    const int m  = lane & 15;
    const int kh = lane >> 4;
    const int d0 = blockIdx.x * 16;
    const int n0 = blockIdx.y * 16;
    v8f acc = {};
    for (int f0 = 0; f0 < D; f0 += 4) {
        v2f a, b;
        a.x = W[(size_t)(f0 + 2 * kh    ) * D + d0 + m];
        a.y = W[(size_t)(f0 + 2 * kh + 1) * D + d0 + m];
        b.x = W[(size_t)(f0 + 2 * kh    ) * D + n0 + m];
        b.y = W[(size_t)(f0 + 2 * kh + 1) * D + n0 + m];
        acc = __builtin_amdgcn_wmma_f32_16x16x4_f32(false, a, false, b,
                                                    (short)0, acc, false, false);
    }
#pragma unroll
    for (int r = 0; r < 8; ++r)
        G[(size_t)(d0 + r + 8 * kh) * D + n0 + m] = acc[r];
}

// ---------------------------------------------------------------------------
// u = W^T b,  bb = b.b   (tiny, one 128-thread block)
// ---------------------------------------------------------------------------
__global__ void k_ub(const float* __restrict__ W, const float* __restrict__ bvec,
                     float* __restrict__ u, float* __restrict__ bb) {
    const int t = threadIdx.x;
    float acc = 0.f;
    for (int f = 0; f < D; ++f) acc += W[(size_t)f * D + t] * bvec[f];
    u[t] = acc;
    if (t == 0) {
        float s = 0.f;
        for (int f = 0; f < D; ++f) s += bvec[f] * bvec[f];
        *bb = s;
    }
}

// ---------------------------------------------------------------------------
// Q = X * G^T + u  (4096x128x128 GEMM), WMMA f32 16x16x4, one wave per tile.
// A[m][kk] = X[b0+m][k0+kk]  (contiguous pair per lane)
// B[kk][n] = G[n0+n][k0+kk]  (contiguous pair per lane)
// ---------------------------------------------------------------------------
__global__ void k_q(const float* __restrict__ X, const float* __restrict__ G,
                    const float* __restrict__ u, float* __restrict__ q) {
    const int lane = threadIdx.x;
    const int m  = lane & 15;
    const int kh = lane >> 4;
    const int b0 = blockIdx.x * 16;
    const int n0 = blockIdx.y * 16;
    v8f acc = {};
    for (int k0 = 0; k0 < D; k0 += 4) {
        v2f a  = *(const v2f*)(X + (size_t)(b0 + m) * D + k0 + 2 * kh);
        v2f bm = *(const v2f*)(G + (size_t)(n0 + m) * D + k0 + 2 * kh);
        acc = __builtin_amdgcn_wmma_f32_16x16x4_f32(false, a, false, bm,
                                                    (short)0, acc, false, false);
    }
    const float un = u[n0 + m];
#pragma unroll
    for (int r = 0; r < 8; ++r)
        q[(size_t)(b0 + r + 8 * kh) * D + n0 + m] = acc[r] + un;
}

// ---------------------------------------------------------------------------
// c[b] = x[b].u + bb
// ---------------------------------------------------------------------------
__global__ void k_c(const float* __restrict__ X, const float* __restrict__ u,
                    const float* __restrict__ bb, float* __restrict__ c) {
    const int b = blockIdx.x * blockDim.x + threadIdx.x;
    if (b >= BATCH) return;
    float acc = *bb;
    for (int d = 0; d < D; ++d) acc += X[(size_t)b * D + d] * u[d];
    c[b] = acc;
}

// ---------------------------------------------------------------------------
// Main memory-bound pass: one block per batch column b, thread e handles edge e.
// e_ij[e] = edge[e,b,:].q[b,:] + c[b];  s = sum_e e_ij;  out = (e_ij/s)*edge.
// Per-lane float4 row reads: 512B contiguous per lane -> full cachelines.
// Second read of the row hits WGP$/L2 (reuse distance = one block phase).
// ---------------------------------------------------------------------------
__global__ void k_out(const float* __restrict__ edge, const float* __restrict__ q,
                      const float* __restrict__ c, float* __restrict__ out) {
    __shared__ float qs[D];
    __shared__ float red[E];
    const int b = blockIdx.x;
    const int e = threadIdx.x;   // 0..127 (E == D == 128)

    qs[e] = q[(size_t)b * D + e];
    __syncthreads();

    const size_t rowoff = ((size_t)e * BATCH + b) * D;
    const float4* row = (const float4*)(edge + rowoff);
    float acc = 0.f;
#pragma unroll 8
    for (int i = 0; i < D / 4; ++i) {
        float4 v = row[i];
        acc += v.x * qs[4 * i] + v.y * qs[4 * i + 1] +
               v.z * qs[4 * i + 2] + v.w * qs[4 * i + 3];
    }
    acc += c[b];

    red[e] = acc;
    __syncthreads();
    for (int off = 64; off > 0; off >>= 1) {
        if (e < off) red[e] += red[e + off];
        __syncthreads();
    }
    const float wt = acc / red[0];

    float4* orow = (float4*)(out + rowoff);
#pragma unroll 8
    for (int i = 0; i < D / 4; ++i) {
        float4 v = row[i];
        orow[i] = make_float4(v.x * wt, v.y * wt, v.z * wt, v.w * wt);
    }
}

extern "C" void kernel_launch(void* const* d_in, const int* in_sizes, int n_in,
                              void* d_out, int out_size, void* d_ws, size_t ws_size,
                              hipStream_t stream) {
    const float* temporal = (const float*)d_in[1];   // [B, D]
    const float* edge     = (const float*)d_in[2];   // [E, B, D]
    const float* W        = (const float*)d_in[3];   // [D, D]
    const float* bvec     = (const float*)d_in[4];   // [D]
    float* out = (float*)d_out;                      // [E, B, D]

    float* ws = (float*)d_ws;
    float* G  = ws + WS_G;
    float* u  = ws + WS_U;
    float* bb = ws + WS_BB;
    float* c  = ws + WS_C;
    float* q  = ws + WS_Q;

    // G = W^T W : 8x8 tiles of 16x16, one wave each
    k_gram<<<dim3(D / 16, D / 16), 32, 0, stream>>>(W, G);
    // u = W^T b, bb = b.b
    k_ub<<<1, D, 0, stream>>>(W, bvec, u, bb);
    // Q = X G^T + u : 256 x 8 tiles, one wave each
    k_q<<<dim3(BATCH / 16, D / 16), 32, 0, stream>>>(temporal, G, u, q);
    // c[b] = x[b].u + bb
    k_c<<<BATCH / 256, 256, 0, stream>>>(temporal, u, bb, c);
    // fused scores + normalization + weighted output
    k_out<<<BATCH, E, 0, stream>>>(edge, q, c, out);
}